// crit_30640296690012
// MI455X (gfx1250) — compile-verified
//
#include <hip/hip_runtime.h>

// CDNA5 / gfx1250, wave32.
typedef __attribute__((ext_vector_type(2))) float v2f;
typedef __attribute__((ext_vector_type(8))) float v8f;

#define L_DIM 18
#define N_DIM 256
#define M_DIM 32000
#define D_DIM 16   // L-2

// Block bt handles time-step t = blockIdx.x (rows 1..16 of input),
// thread b = threadIdx.x handles batch column b. One gather per thread.
// Per-block reduction via V_WMMA_F32_16X16X4_F32 (full fp32 matrix path):
//   D = A(16x4) * ones(4x16) + 0  =>  sum(all 256 D slots) == 16 * sum(all A slots)
// This identity holds for ANY lane<->element striping, so it is robust to
// VGPR-layout details while keeping the reduction exact-order deterministic.
__global__ __launch_bounds__(256) void ce_gather_partial_kernel(
    const float* __restrict__ input,   // [L, N, M] fp32 log-probs
    const int*   __restrict__ seq,     // [D, N] token ids
    float*       __restrict__ ws)      // [2*D] per-block partials (16*S, 16*C)
{
    const int t = blockIdx.x;    // 0..D-1
    const int b = threadIdx.x;   // 0..N-1

    // --- masked-gather address generation -------------------------------
    int tok = seq[t * N_DIM + b];
    bool isz = (tok == 0);
    bool prev_zero = false;
    for (int j = 0; j < t; ++j) {               // <=15 coalesced L2-hot loads
        prev_zero = prev_zero || (seq[j * N_DIM + b] == 0);
    }
    const bool firstz  = isz && !prev_zero;     // first zero -> END token
    const bool contrib = (!isz) || firstz;
    const int  target  = firstz ? (M_DIM - 1) : tok;

    float val = 0.0f;
    if (contrib) {
        // rows 1..D of input: input[t+1, b, target]
        long long idx = ((long long)(t + 1) * N_DIM + b) * (long long)M_DIM
                        + (long long)target;
        val = input[idx];                       // 1 scattered global_load_b32
    }
    const float cnt = contrib ? 1.0f : 0.0f;

    // --- fp32 WMMA reduction (uniform across all 8 waves, EXEC all-1s) ---
    v2f a_s; a_s.x = val; a_s.y = 0.0f;
    v2f a_c; a_c.x = cnt; a_c.y = 0.0f;
    v2f ones; ones.x = 1.0f; ones.y = 1.0f;
    v8f zero = {};
    // (neg_a, A, neg_b, B, c_mod, C, reuse_a, reuse_b)
    v8f ds = __builtin_amdgcn_wmma_f32_16x16x4_f32(
        false, a_s, false, ones, (short)0, zero, false, false);
    v8f dc = __builtin_amdgcn_wmma_f32_16x16x4_f32(
        false, a_c, false, ones, (short)0, zero, false, false);

    float d8s = ds[0] + ds[1] + ds[2] + ds[3] + ds[4] + ds[5] + ds[6] + ds[7];
    float d8c = dc[0] + dc[1] + dc[2] + dc[3] + dc[4] + dc[5] + dc[6] + dc[7];

    // Per-wave: sum over its 32 lanes of d8 == 16 * (wave's partial total).
    // Fold all 256 lanes in a fixed order (deterministic, no float atomics).
    __shared__ float red_s[256];
    __shared__ float red_c[256];
    red_s[b] = d8s;
    red_c[b] = d8c;
    __syncthreads();

    if (b == 0) {
        float S = 0.0f, C = 0.0f;
        for (int i = 0; i < 256; ++i) { S += red_s[i]; C += red_c[i]; }
        ws[2 * t]     = S;   // == 16 * block sum   (x16 is exact fp scaling)
        ws[2 * t + 1] = C;   // == 16 * block count (small ints: exact)
    }
}

__global__ void ce_finalize_kernel(const float* __restrict__ ws,
                                   float* __restrict__ out)
{
    float S = 0.0f, C = 0.0f;
    for (int i = 0; i < D_DIM; ++i) {
        S += ws[2 * i];
        C += ws[2 * i + 1];
    }
    // -(16*sum)/(16*count) == -sum/count ; the WMMA x16 factors cancel.
    out[0] = -S / C;
}

extern "C" void kernel_launch(void* const* d_in, const int* in_sizes, int n_in,
                              void* d_out, int out_size, void* d_ws, size_t ws_size,
                              hipStream_t stream)
{
    (void)in_sizes; (void)n_in; (void)out_size; (void)ws_size;
    const float* input = (const float*)d_in[0];   // [18,256,32000] fp32
    const int*   seq   = (const int*)d_in[1];     // [16,256] int
    float*       out   = (float*)d_out;           // scalar fp32 loss
    float*       ws    = (float*)d_ws;            // 2*D floats of scratch

    ce_gather_partial_kernel<<<dim3(D_DIM), dim3(N_DIM), 0, stream>>>(input, seq, ws);
    ce_finalize_kernel<<<dim3(1), dim3(1), 0, stream>>>(ws, out);
}